// PoolVerticesToEdges_1529008357761
// MI455X (gfx1250) — compile-verified
//
#include <hip/hip_runtime.h>

// PoolVerticesToEdges:  e_out[E,D] = e_in + conEd^T[E,V] @ v_in[V,D]
// V=4096, E=16384, D=256, all fp32.
//
// GEMM view: C[m,n] = sum_k A[m,k]*B[k,n],  A[m,k]=conEd[k*E+m], B[k,n]=v_in[k*D+n]
// M=16384 (E), N=256 (D), K=4096 (V).
//
// fp32 WMMA 16x16x4, wave32.  Fragment layouts (CDNA5 ISA 7.12.2):
//   A (16x4 f32, 2 VGPRs): lane holds M = lane&15; v0 -> K=2*(lane>>4), v1 -> K=2*(lane>>4)+1
//   B (4x16 f32, 2 VGPRs): lane holds N = lane&15; v0 -> K=2*(lane>>4), v1 -> K=2*(lane>>4)+1
//   C/D (16x16 f32, 8 VGPRs): c[g] = C[g + 8*(lane>>4)][lane&15]

typedef __attribute__((ext_vector_type(2))) float v2f;
typedef __attribute__((ext_vector_type(8))) float v8f;

#define V_DIM 4096
#define E_DIM 16384
#define D_DIM 256

#define BLOCK_M 128   // per workgroup (2 wave-rows of 64)
#define WAVE_M  64    // 4 x 16 M-subtiles per wave
#define WAVE_N  64    // 4 x 16 N-subtiles per wave

__global__ __launch_bounds__(256, 1)
void pool_vertices_to_edges_wmma(const float* __restrict__ vin,   // [V, D]
                                 const float* __restrict__ ein,   // [E, D]
                                 const float* __restrict__ conEd, // [V, E]
                                 float* __restrict__ out)         // [E, D]
{
    const int lane = threadIdx.x & 31;
    const int wave = threadIdx.x >> 5;
    const int lo   = lane & 15;   // M (for A) / N (for B) index within tile
    const int hi   = lane >> 4;   // selects K pair {2*hi, 2*hi+1}

    // 8 waves: 2 along M x 4 along N  -> block tile 128M x 256N
    const int mWave = blockIdx.x * BLOCK_M + (wave >> 2) * WAVE_M;
    const int nWave = (wave & 3) * WAVE_N;

    v8f acc[4][4];
#pragma unroll
    for (int i = 0; i < 4; ++i)
#pragma unroll
        for (int j = 0; j < 4; ++j)
            acc[i][j] = v8f{0.f, 0.f, 0.f, 0.f, 0.f, 0.f, 0.f, 0.f};

    // Per-lane streaming pointers.
    // A frag i, reg0: conEd[(k + 2*hi    )*E + mWave + 16*i + lo]
    //           reg1: conEd[(k + 2*hi + 1)*E + ...]            (advance 4*E per K-step)
    const float* aBase = conEd + (size_t)(2 * hi) * E_DIM + (mWave + lo);
    // B frag j, reg0: vin[(k + 2*hi)*D + nWave + 16*j + lo]     (advance 4*D per K-step)
    const float* bBase = vin + (size_t)(2 * hi) * D_DIM + (nWave + lo);

    for (int k = 0; k < V_DIM; k += 4) {
        // Prefetch the conEd stream 64 K-rows (4 MB) ahead; speculative,
        // silently dropped past the end of the buffer.
        __builtin_prefetch(aBase + (size_t)64 * E_DIM, 0, 0);

        v2f a[4], b[4];
#pragma unroll
        for (int i = 0; i < 4; ++i) {
            a[i].x = aBase[16 * i];
            a[i].y = aBase[16 * i + E_DIM];
        }
#pragma unroll
        for (int j = 0; j < 4; ++j) {
            b[j].x = bBase[16 * j];
            b[j].y = bBase[16 * j + D_DIM];
        }

#pragma unroll
        for (int i = 0; i < 4; ++i)
#pragma unroll
            for (int j = 0; j < 4; ++j)
                acc[i][j] = __builtin_amdgcn_wmma_f32_16x16x4_f32(
                    /*neg_a=*/false, a[i],
                    /*neg_b=*/false, b[j],
                    /*c_mod=*/(short)0, acc[i][j],
                    /*reuse_a=*/false, /*reuse_b=*/false);

        aBase += (size_t)4 * E_DIM;
        bBase += (size_t)4 * D_DIM;
    }

    // Epilogue: residual add + store.  c[g] = C[m0+g][n0], rows coalesced over 16 lanes.
#pragma unroll
    for (int i = 0; i < 4; ++i) {
#pragma unroll
        for (int j = 0; j < 4; ++j) {
            const int m0 = mWave + 16 * i + 8 * hi;
            const int n0 = nWave + 16 * j + lo;
#pragma unroll
            for (int g = 0; g < 8; ++g) {
                const size_t idx = (size_t)(m0 + g) * D_DIM + n0;
                out[idx] = ein[idx] + acc[i][j][g];
            }
        }
    }
}

extern "C" void kernel_launch(void* const* d_in, const int* in_sizes, int n_in,
                              void* d_out, int out_size, void* d_ws, size_t ws_size,
                              hipStream_t stream) {
    // setup_inputs order: v_in, e_in, u_in, adj, conEd
    const float* vin   = (const float*)d_in[0];
    const float* ein   = (const float*)d_in[1];
    const float* conEd = (const float*)d_in[4];
    float* out = (float*)d_out;

    dim3 grid(E_DIM / BLOCK_M);  // 128 blocks
    dim3 block(256);             // 8 wave32s
    hipLaunchKernelGGL(pool_vertices_to_edges_wmma, grid, block, 0, stream,
                       vin, ein, conEd, out);
}